// DMMHLoss_2353642078712
// MI455X (gfx1250) — compile-verified
//
#include <hip/hip_runtime.h>

// ---------------------------------------------------------------------------
// DMMH loss on MI455X (gfx1250). f32 WMMA (V_WMMA_F32_16X16X4_F32) for the
// Gram matrix u@u^T; mask-sparse triple loss; deterministic reductions.
// ---------------------------------------------------------------------------

#define NPTS 384
#define BITK 64
#define NCLS 100
#define LAMBDA 1e-4f
#define PI_OVER_256 0.01227184630308512983774470071594f  // pi/(4*k), k=64

typedef float v2f __attribute__((ext_vector_type(2)));
typedef float v8f __attribute__((ext_vector_type(8)));

// ---------------------------------------------------------------------------
// Kernel 1: one wave (32 lanes) per 16x16 tile of G = u @ u^T, K=64 swept in
// 16 steps of V_WMMA_F32_16X16X4_F32. Fuses c = cos((G + 64) * pi/256).
//
// 16x4 f32 A layout: lane l (l<16) holds A[l][k0+0..1]; lane l (>=16) holds
// A[l-16][k0+2..3]  -> contiguous float2 load per lane. B (4x16 = u^T tile)
// mirrors this, so both operands are float2 row loads from u.
// C/D layout: VGPR v, lane l -> element (M = v + 8*(l>>4), N = l&15).
// ---------------------------------------------------------------------------
__global__ __launch_bounds__(32) void gram_cos_kernel(const float* __restrict__ u,
                                                      float* __restrict__ cmat) {
  const int tile = blockIdx.x;          // 0 .. 24*24-1
  const int rt   = tile / (NPTS / 16);
  const int ct   = tile % (NPTS / 16);
  const int lane = threadIdx.x;         // 0..31
  const int half = lane >> 4;           // 0 or 1
  const int l15  = lane & 15;
  const int rowA = rt * 16 + l15;       // row of u for the A tile
  const int rowB = ct * 16 + l15;       // row of u for the B (= u^T) tile
  const int koff = 2 * half;            // K sub-offset within the 4-wide step

  v8f acc = {0.f, 0.f, 0.f, 0.f, 0.f, 0.f, 0.f, 0.f};

#pragma unroll
  for (int k0 = 0; k0 < BITK; k0 += 4) {
    v2f a = *(const v2f*)(u + rowA * BITK + k0 + koff);
    v2f b = *(const v2f*)(u + rowB * BITK + k0 + koff);
    // D = A x B + C   (8 args: neg_a, A, neg_b, B, c_mod, C, reuse_a, reuse_b)
    acc = __builtin_amdgcn_wmma_f32_16x16x4_f32(false, a, false, b,
                                                (short)0, acc, false, false);
  }

#pragma unroll
  for (int v = 0; v < 8; ++v) {
    const int m = rt * 16 + v + 8 * half;
    const int n = ct * 16 + l15;
    cmat[m * NPTS + n] = cosf((acc[v] + (float)BITK) * PI_OVER_256);
  }
}

// ---------------------------------------------------------------------------
// Kernel 2: labels[i] = argmax_c y[i][c]  (y is one-hot f32)
// ---------------------------------------------------------------------------
__global__ __launch_bounds__(256) void labels_kernel(const float* __restrict__ y,
                                                     int* __restrict__ labels) {
  const int i = blockIdx.x * blockDim.x + threadIdx.x;
  if (i < NPTS) {
    int lab = 0;
    float best = -1.0f;
    for (int c = 0; c < NCLS; ++c) {
      const float v = y[i * NCLS + c];
      if (v > best) { best = v; lab = c; }
    }
    labels[i] = lab;
  }
}

__device__ __forceinline__ float log_sigmoid_f(float x) {
  // stable: min(x,0) - log1p(exp(-|x|))
  return fminf(x, 0.0f) - log1pf(__expf(-fabsf(x)));
}

// ---------------------------------------------------------------------------
// Kernel 3: one block per row r. Collect positives (label[j]==label[r]) into
// LDS deterministically, then sum log_sigmoid(6*(c[r,neg] - c[r,pos])) over
// the neg x pos pairs. Tree-reduce; emit rowLoss[r], rowValid[r].
// ---------------------------------------------------------------------------
__global__ __launch_bounds__(256) void row_loss_kernel(const float* __restrict__ cmat,
                                                       const int* __restrict__ labels,
                                                       float* __restrict__ rowLoss,
                                                       float* __restrict__ rowValid) {
  __shared__ int   posList[NPTS];
  __shared__ float posC[NPTS];
  __shared__ int   s_npos;
  __shared__ float red[256];

  const int r   = blockIdx.x;
  const int tid = threadIdx.x;
  const int myLab = labels[r];

  if (tid == 0) {                       // serial build -> deterministic order
    int cnt = 0;
    for (int j = 0; j < NPTS; ++j)
      if (labels[j] == myLab) posList[cnt++] = j;
    s_npos = cnt;
  }
  __syncthreads();

  const int npos = s_npos;
  for (int j = tid; j < npos; j += blockDim.x)
    posC[j] = cmat[r * NPTS + posList[j]];
  __syncthreads();

  float sum = 0.0f;
  for (int i = tid; i < NPTS; i += blockDim.x) {
    if (labels[i] != myLab) {           // i is a negative for row r
      const float ci = cmat[r * NPTS + i];
      for (int j = 0; j < npos; ++j)
        sum += log_sigmoid_f(6.0f * (ci - posC[j]));
    }
  }

  red[tid] = sum;
  __syncthreads();
  for (int s = 128; s > 0; s >>= 1) {
    if (tid < s) red[tid] += red[tid + s];
    __syncthreads();
  }

  if (tid == 0) {
    const int nneg = NPTS - npos;
    const float pairs = (float)(npos * nneg);
    const bool valid = pairs > 0.0f;
    const float rl = -red[0] / fmaxf(pairs, 1.0f);
    rowLoss[r]  = valid ? rl : 0.0f;
    rowValid[r] = valid ? 1.0f : 0.0f;
  }
}

// ---------------------------------------------------------------------------
// Kernel 4: single block. loss1 = mean(rowLoss over valid rows);
// loss2 = LAMBDA * mean((u - sign(u))^2). out[0] = loss1 + loss2.
// ---------------------------------------------------------------------------
__global__ __launch_bounds__(256) void final_kernel(const float* __restrict__ rowLoss,
                                                    const float* __restrict__ rowValid,
                                                    const float* __restrict__ u,
                                                    float* __restrict__ out) {
  __shared__ float redA[256], redB[256], redC[256];
  const int tid = threadIdx.x;

  float sL = 0.0f, sV = 0.0f, sQ = 0.0f;
  for (int r = tid; r < NPTS; r += 256) { sL += rowLoss[r]; sV += rowValid[r]; }
  for (int i = tid; i < NPTS * BITK; i += 256) {
    const float x  = u[i];
    const float sg = (x > 0.0f) ? 1.0f : ((x < 0.0f) ? -1.0f : 0.0f);
    const float d  = x - sg;
    sQ += d * d;
  }

  redA[tid] = sL; redB[tid] = sV; redC[tid] = sQ;
  __syncthreads();
  for (int s = 128; s > 0; s >>= 1) {
    if (tid < s) { redA[tid] += redA[tid+s]; redB[tid] += redB[tid+s]; redC[tid] += redC[tid+s]; }
    __syncthreads();
  }

  if (tid == 0) {
    const float count = redB[0];
    const float loss1 = (count > 0.0f) ? (redA[0] / fmaxf(count, 1.0f)) : 0.0f;
    const float loss2 = LAMBDA * (redC[0] / (float)(NPTS * BITK));
    out[0] = loss1 + loss2;
  }
}

// ---------------------------------------------------------------------------
extern "C" void kernel_launch(void* const* d_in, const int* in_sizes, int n_in,
                              void* d_out, int out_size, void* d_ws, size_t ws_size,
                              hipStream_t stream) {
  const float* u = (const float*)d_in[0];   // [384, 64]  f32
  const float* y = (const float*)d_in[1];   // [384, 100] f32
  float* out = (float*)d_out;               // scalar f32

  // Workspace layout
  char* ws = (char*)d_ws;
  float* cmat     = (float*)ws;                                   // 384*384 f32
  int*   labels   = (int*)  (ws + (size_t)NPTS * NPTS * 4);       // 384 i32
  float* rowLoss  = (float*)(ws + (size_t)NPTS * NPTS * 4 + NPTS * 4);
  float* rowValid = (float*)(ws + (size_t)NPTS * NPTS * 4 + NPTS * 8);

  const int tiles = (NPTS / 16) * (NPTS / 16);  // 576 tiles, 1 wave each
  gram_cos_kernel<<<dim3(tiles), dim3(32), 0, stream>>>(u, cmat);
  labels_kernel<<<dim3((NPTS + 255) / 256), dim3(256), 0, stream>>>(y, labels);
  row_loss_kernel<<<dim3(NPTS), dim3(256), 0, stream>>>(cmat, labels, rowLoss, rowValid);
  final_kernel<<<dim3(1), dim3(256), 0, stream>>>(rowLoss, rowValid, u, out);
}